// Attention_27041114096331
// MI455X (gfx1250) — compile-verified
//
#include <hip/hip_runtime.h>
#include <hip/hip_bf16.h>
#include <math.h>

typedef __attribute__((ext_vector_type(2))) float v2f;
typedef __attribute__((ext_vector_type(8))) float v8f;

#define B_DIM   64
#define T_DIM   1000
#define EMB     512
#define ATT     128
#define NF      32
#define KCONV   31
#define ECHUNK  64
#define LPAD    132   // LDS row stride (128 + 4) floats; 528 B = 16-byte aligned

// Async global->LDS copy of 16 bytes per lane (CDNA5, tracked by ASYNCcnt).
// ldsoff = wave-relative LDS byte offset (low 32 bits of generic LDS pointer).
__device__ __forceinline__ void async_copy_b128(unsigned ldsoff, const float* gptr)
{
    asm volatile("global_load_async_to_lds_b128 %0, %1, off"
                 :: "v"(ldsoff), "v"(gptr) : "memory");
}
__device__ __forceinline__ void async_wait0()
{
    asm volatile("s_wait_asynccnt 0" ::: "memory");
}

// ---------------------------------------------------------------------------
// Kernel 1: conv1d (SAME, width 31, in_ch 2, out_ch 32) + eval-mode batchnorm.
// Output layout convbn[b][t][f]  (f contiguous -> WMMA A-fragments are b64).
// ---------------------------------------------------------------------------
__global__ __launch_bounds__(256)
void convbn_kernel(const float* __restrict__ aw,     // [B,2,T]
                   const float* __restrict__ cw,     // [NF,2,K]
                   const float* __restrict__ cb,     // [NF]
                   const float* __restrict__ gamma_,
                   const float* __restrict__ beta_,
                   const float* __restrict__ mean_,
                   const float* __restrict__ var_,
                   float* __restrict__ convbn)       // [B,T,NF]
{
    int gid = blockIdx.x * blockDim.x + threadIdx.x;
    if (gid >= B_DIM * T_DIM * NF) return;
    int f = gid & (NF - 1);
    int t = (gid / NF) % T_DIM;
    int b = gid / (NF * T_DIM);

    float s = cb[f];
    #pragma unroll
    for (int c = 0; c < 2; ++c) {
        const float* x = aw + ((size_t)b * 2 + c) * T_DIM;
        const float* w = cw + ((size_t)f * 2 + c) * KCONV;
        #pragma unroll
        for (int k = 0; k < KCONV; ++k) {
            int tt = t + k - (KCONV / 2);
            float xv = (tt >= 0 && tt < T_DIM) ? x[tt] : 0.0f;
            s = fmaf(xv, w[k], s);
        }
    }
    float inv = rsqrtf(var_[f] + 1e-5f);
    s = (s - mean_[f]) * (gamma_[f] * inv) + beta_[f];
    convbn[gid] = s;
}

// ---------------------------------------------------------------------------
// Kernel 2: q = query @ Wq   (64x1024 @ 1024x128) via WMMA f32 16x16x4.
// One wave per 16x16 output tile. grid = 4 M-tiles * 8 N-tiles = 32 blocks.
// ---------------------------------------------------------------------------
__global__ __launch_bounds__(32)
void qproj_kernel(const float* __restrict__ query,   // [64,1024]
                  const float* __restrict__ Wq,      // [1024,128]
                  float* __restrict__ qout)          // [64,128]
{
    const int lane   = threadIdx.x & 31;
    const int lane16 = lane & 15;
    const int khalf  = lane >> 4;
    const int mt = blockIdx.x >> 3;   // 0..3
    const int nt = blockIdx.x & 7;    // 0..7
    const int row = mt * 16 + lane16;

    v8f acc = {};
    for (int k0 = 0; k0 < 1024; k0 += 4) {
        v2f a = *(const v2f*)(query + (size_t)row * 1024 + k0 + 2 * khalf);
        v2f bf;
        bf[0] = Wq[(size_t)(k0 + 2 * khalf    ) * ATT + nt * 16 + lane16];
        bf[1] = Wq[(size_t)(k0 + 2 * khalf + 1) * ATT + nt * 16 + lane16];
        acc = __builtin_amdgcn_wmma_f32_16x16x4_f32(false, a, false, bf,
                                                    (short)0, acc, false, false);
    }
    #pragma unroll
    for (int r = 0; r < 8; ++r)
        qout[(size_t)(mt * 16 + r + 8 * khalf) * ATT + nt * 16 + lane16] = acc[r];
}

// ---------------------------------------------------------------------------
// Kernel 3: fused  energies[b,t] = tanh(q + key@Wk + convbn@Wloc) . v
// Block = 128 threads (4 waves). Block handles (b, 64 t-rows); each wave a
// 16x128 tile in 8 f32 accumulators. Wk chunks + Wloc staged in LDS via
// GLOBAL_LOAD_ASYNC_TO_LDS_B128 (ASYNCcnt).
// ---------------------------------------------------------------------------
__global__ __launch_bounds__(128)
void energies_kernel(const float* __restrict__ key,     // [B,T,EMB]
                     const float* __restrict__ Wk,      // [EMB,ATT]
                     const float* __restrict__ Wloc,    // [NF,ATT]
                     const float* __restrict__ convbn,  // [B,T,NF]
                     const float* __restrict__ qproj,   // [B,ATT]
                     const float* __restrict__ vvec,    // [ATT]
                     float* __restrict__ energies)      // [B,T]
{
    __shared__ __align__(16) float ldsW[ECHUNK * LPAD];  // ~33.8 KB
    __shared__ __align__(16) float ldsL[NF * LPAD];      // ~16.9 KB

    const int tid    = threadIdx.x;
    const int lane   = tid & 31;
    const int waveid = tid >> 5;
    const int lane16 = lane & 15;
    const int khalf  = lane >> 4;
    const int b  = blockIdx.y;
    const int t0 = blockIdx.x * 64 + waveid * 16;
    int trow = t0 + lane16;
    if (trow > T_DIM - 1) trow = T_DIM - 1;   // clamp (rows >= T masked on store)

    // ---- stage Wloc (32x128) into padded LDS, async b128 (8 per thread) ----
    {
        #pragma unroll
        for (int i = 0; i < (NF * ATT / 4) / 128; ++i) {
            int idx4 = tid + i * 128;
            int r  = idx4 >> 5;          // 32 float4 per 128-float row
            int c4 = idx4 & 31;
            unsigned ldsoff = (unsigned)(uintptr_t)&ldsL[r * LPAD + c4 * 4];
            async_copy_b128(ldsoff, Wloc + (size_t)r * ATT + c4 * 4);
        }
        async_wait0();
    }
    __syncthreads();

    v8f acc[8] = {};

    // ---- location term: convbn[b, t0..t0+15, 0..31] @ Wloc ----
    const float* convRow = convbn + ((size_t)b * T_DIM + trow) * NF;
    #pragma unroll
    for (int kk = 0; kk < NF; kk += 4) {
        v2f a = *(const v2f*)(convRow + kk + 2 * khalf);
        #pragma unroll
        for (int nt = 0; nt < 8; ++nt) {
            v2f bf;
            bf[0] = ldsL[(kk + 2 * khalf    ) * LPAD + nt * 16 + lane16];
            bf[1] = ldsL[(kk + 2 * khalf + 1) * LPAD + nt * 16 + lane16];
            acc[nt] = __builtin_amdgcn_wmma_f32_16x16x4_f32(false, a, false, bf,
                                                            (short)0, acc[nt], false, false);
        }
    }

    // ---- key term: key[b, t-tile, :] @ Wk, K chunked through LDS ----
    const float* keyRow = key + ((size_t)b * T_DIM + trow) * EMB;
    for (int ec = 0; ec < EMB; ec += ECHUNK) {
        __syncthreads();   // previous chunk fully consumed
        // async-stage Wk[ec..ec+63][0..127] into padded LDS (16 b128 / thread)
        #pragma unroll
        for (int i = 0; i < (ECHUNK * ATT / 4) / 128; ++i) {
            int idx4 = tid + i * 128;
            int r  = idx4 >> 5;
            int c4 = idx4 & 31;
            unsigned ldsoff = (unsigned)(uintptr_t)&ldsW[r * LPAD + c4 * 4];
            async_copy_b128(ldsoff, Wk + (size_t)(ec + r) * ATT + c4 * 4);
        }
        async_wait0();
        __syncthreads();

        if (ec + ECHUNK < EMB)
            __builtin_prefetch(keyRow + ec + ECHUNK + 2 * khalf, 0, 1);

        #pragma unroll
        for (int kk = 0; kk < ECHUNK; kk += 4) {
            v2f a = *(const v2f*)(keyRow + ec + kk + 2 * khalf);
            #pragma unroll
            for (int nt = 0; nt < 8; ++nt) {
                v2f bf;
                bf[0] = ldsW[(kk + 2 * khalf    ) * LPAD + nt * 16 + lane16];
                bf[1] = ldsW[(kk + 2 * khalf + 1) * LPAD + nt * 16 + lane16];
                acc[nt] = __builtin_amdgcn_wmma_f32_16x16x4_f32(false, a, false, bf,
                                                                (short)0, acc[nt], false, false);
            }
        }
    }

    // ---- epilogue: e[row] = sum_n tanh(acc + q[b,n]) * v[n] ----
    float part[8] = {0, 0, 0, 0, 0, 0, 0, 0};
    #pragma unroll
    for (int nt = 0; nt < 8; ++nt) {
        float qf = qproj[(size_t)b * ATT + nt * 16 + lane16];
        float vf = vvec[nt * 16 + lane16];
        #pragma unroll
        for (int r = 0; r < 8; ++r)
            part[r] = fmaf(tanhf(acc[nt][r] + qf), vf, part[r]);
    }
    // reduce across the 16 lanes of each half-wave (C layout: M = r + 8*khalf)
    #pragma unroll
    for (int r = 0; r < 8; ++r) {
        part[r] += __shfl_xor(part[r], 1, 32);
        part[r] += __shfl_xor(part[r], 2, 32);
        part[r] += __shfl_xor(part[r], 4, 32);
        part[r] += __shfl_xor(part[r], 8, 32);
    }
    if (lane16 == 0) {          // lanes 0 (rows 0..7) and 16 (rows 8..15)
        #pragma unroll
        for (int r = 0; r < 8; ++r) {
            int to = t0 + 8 * khalf + r;
            if (to < T_DIM)
                energies[(size_t)b * T_DIM + to] = part[r];
        }
    }
}

// ---------------------------------------------------------------------------
// Kernel 4: masked softmax over T per batch row.
// ---------------------------------------------------------------------------
__global__ __launch_bounds__(256)
void softmax_kernel(const float* __restrict__ energies,
                    const unsigned char* __restrict__ mask,
                    float* __restrict__ wout)           // [B,T]
{
    __shared__ float red[256];
    const int b = blockIdx.x, tid = threadIdx.x;
    const float* eb = energies + (size_t)b * T_DIM;
    const unsigned char* mb = mask + (size_t)b * T_DIM;

    float m = -INFINITY;
    for (int t = tid; t < T_DIM; t += 256) {
        float e = mb[t] ? -INFINITY : eb[t];
        m = fmaxf(m, e);
    }
    red[tid] = m; __syncthreads();
    for (int s = 128; s > 0; s >>= 1) {
        if (tid < s) red[tid] = fmaxf(red[tid], red[tid + s]);
        __syncthreads();
    }
    m = red[0]; __syncthreads();

    float sum = 0.0f;
    for (int t = tid; t < T_DIM; t += 256) {
        float e = mb[t] ? -INFINITY : eb[t];
        sum += __expf(e - m);
    }
    red[tid] = sum; __syncthreads();
    for (int s = 128; s > 0; s >>= 1) {
        if (tid < s) red[tid] += red[tid + s];
        __syncthreads();
    }
    float inv = 1.0f / red[0];

    for (int t = tid; t < T_DIM; t += 256) {
        float e = mb[t] ? -INFINITY : eb[t];
        wout[(size_t)b * T_DIM + t] = __expf(e - m) * inv;
    }
}

// ---------------------------------------------------------------------------
// Kernel 5: context[b,e] = sum_t w[b,t] * key[b,t,e]   (bandwidth-bound)
// ---------------------------------------------------------------------------
__global__ __launch_bounds__(256)
void context_kernel(const float* __restrict__ key,  // [B,T,EMB]
                    const float* __restrict__ w,    // [B,T]
                    float* __restrict__ ctx)        // [B,EMB]
{
    const int b = blockIdx.y;
    const int e = blockIdx.x * 256 + threadIdx.x;
    const float* kb = key + (size_t)b * T_DIM * EMB;
    const float* wb = w + (size_t)b * T_DIM;
    float s = 0.0f;
    for (int t = 0; t < T_DIM; ++t)
        s = fmaf(wb[t], kb[(size_t)t * EMB + e], s);
    ctx[(size_t)b * EMB + e] = s;
}

// ---------------------------------------------------------------------------
extern "C" void kernel_launch(void* const* d_in, const int* in_sizes, int n_in,
                              void* d_out, int out_size, void* d_ws, size_t ws_size,
                              hipStream_t stream)
{
    const float* query = (const float*)d_in[0];
    const float* key   = (const float*)d_in[1];
    const float* awcat = (const float*)d_in[2];
    const unsigned char* mask = (const unsigned char*)d_in[3];
    const float* Wq    = (const float*)d_in[4];
    const float* Wk    = (const float*)d_in[5];
    const float* convw = (const float*)d_in[6];
    const float* convb = (const float*)d_in[7];
    const float* gamma_ = (const float*)d_in[8];
    const float* beta_  = (const float*)d_in[9];
    const float* mean_  = (const float*)d_in[10];
    const float* var_   = (const float*)d_in[11];
    const float* Wloc  = (const float*)d_in[12];
    const float* vvec  = (const float*)d_in[13];

    float* out = (float*)d_out;
    float* ctx     = out;                       // [64,512]
    float* weights = out + B_DIM * EMB;         // [64,1000]

    float* wsf = (float*)d_ws;
    float* convbn   = wsf;                                    // 2,048,000 f
    float* qproj    = wsf + (size_t)B_DIM * T_DIM * NF;       // 8,192 f
    float* energies = qproj + (size_t)B_DIM * ATT;            // 64,000 f

    // 1) conv1d + BN
    {
        int n = B_DIM * T_DIM * NF;
        convbn_kernel<<<(n + 255) / 256, 256, 0, stream>>>(
            awcat, convw, convb, gamma_, beta_, mean_, var_, convbn);
    }
    // 2) query projection
    qproj_kernel<<<32, 32, 0, stream>>>(query, Wq, qproj);

    // 3) fused energies (WMMA + async-to-LDS staging)
    {
        dim3 grid((T_DIM + 63) / 64, B_DIM);
        energies_kernel<<<grid, 128, 0, stream>>>(
            key, Wk, Wloc, convbn, qproj, vvec, energies);
    }
    // 4) softmax -> attention weights (second output region)
    softmax_kernel<<<B_DIM, 256, 0, stream>>>(energies, mask, weights);

    // 5) context = weights @ key (first output region)
    {
        dim3 grid(EMB / 256, B_DIM);
        context_kernel<<<grid, 256, 0, stream>>>(key, weights, ctx);
    }
}